// SimGNN_84585085927778
// MI455X (gfx1250) — compile-verified
//
#include <hip/hip_runtime.h>
#include <math.h>

typedef __attribute__((ext_vector_type(2))) float v2f;
typedef __attribute__((ext_vector_type(8))) float v8f;

// ---------------------------------------------------------------------------
// Utility kernels
// ---------------------------------------------------------------------------
__global__ void zero_f(float* __restrict__ p, int n) {
    int t = blockIdx.x * blockDim.x + threadIdx.x;
    if (t < n) p[t] = 0.0f;
}

__global__ void deg_count(const int* __restrict__ dst, float* __restrict__ deg, int E) {
    int t = blockIdx.x * blockDim.x + threadIdx.x;
    if (t < E) atomicAdd(&deg[dst[t]], 1.0f);
}

__global__ void finish_dinv(float* __restrict__ deg, int N) {
    int t = blockIdx.x * blockDim.x + threadIdx.x;
    if (t < N) deg[t] = rsqrtf(deg[t] + 1.0f);
}

__global__ void relu_k(float* __restrict__ p, int n) {
    int t = blockIdx.x * blockDim.x + threadIdx.x;
    if (t < n) p[t] = fmaxf(p[t], 0.0f);
}

// ---------------------------------------------------------------------------
// fp32 WMMA GEMM: H[N x Cout] = X[N x Cin] @ W[Cin x Cout]
// One wave (32 lanes) computes one 16x16 output tile with V_WMMA_F32_16X16X4_F32.
// ISA layouts (CDNA5 ISA §7.12.2):
//   A 16x4  : lane L -> row M=L&15 ; VGPR0=K(2*(L>>4)) , VGPR1=K(2*(L>>4)+1)
//   B 4x16  : lane L -> col N=L&15 ; VGPR0=K(2*(L>>4)) , VGPR1=K(2*(L>>4)+1)
//   C/D 16x16: VGPR r, lane L -> M = r + 8*(L>>4), N = L&15
// ---------------------------------------------------------------------------
__global__ __launch_bounds__(256)
void gemm_wmma_f32(const float* __restrict__ X, const float* __restrict__ W,
                   float* __restrict__ H, int N, int Cin, int Cout) {
    int wavesPerBlk = blockDim.x >> 5;
    int wave  = blockIdx.x * wavesPerBlk + (threadIdx.x >> 5);
    int tilesN = Cout >> 4;
    int tileM = wave / tilesN;
    int tileN = wave % tilesN;
    if (tileM * 16 >= N) return;               // whole wave exits together (EXEC all-1 otherwise)

    int lane = threadIdx.x & 31;
    int half = lane >> 4;                      // 0 or 1
    int idx  = lane & 15;

    v8f acc = {};
    const float* xrow = X + (size_t)(tileM * 16 + idx) * Cin;
    const float* wcol = W + (size_t)(tileN * 16 + idx);

    for (int k = 0; k < Cin; k += 4) {
        int ka = k + half * 2;
        v2f a, b;
        a.x = xrow[ka];
        a.y = xrow[ka + 1];
        b.x = wcol[(size_t)ka * Cout];
        b.y = wcol[(size_t)(ka + 1) * Cout];
        acc = __builtin_amdgcn_wmma_f32_16x16x4_f32(
            /*neg_a=*/false, a, /*neg_b=*/false, b,
            /*c_mod=*/(short)0, acc, /*reuse_a=*/false, /*reuse_b=*/false);
    }

#pragma unroll
    for (int r = 0; r < 8; ++r) {
        H[(size_t)(tileM * 16 + r + half * 8) * Cout + tileN * 16 + idx] = acc[r];
    }
}

// ---------------------------------------------------------------------------
// GCN aggregation: A = H*dinv^2 + b   (self-loop term + bias), then edge scatter
// ---------------------------------------------------------------------------
__global__ void init_agg(const float* __restrict__ H, const float* __restrict__ dinv,
                         const float* __restrict__ bias, float* __restrict__ A,
                         int N, int C) {
    int t = blockIdx.x * blockDim.x + threadIdx.x;
    if (t >= N * C) return;
    int i = t / C, c = t - i * C;
    float di = dinv[i];
    A[t] = H[t] * di * di + bias[c];
}

__global__ void edge_agg(const int* __restrict__ src, const int* __restrict__ dst,
                         const float* __restrict__ H, const float* __restrict__ dinv,
                         float* __restrict__ A, int E, int C) {
    int t = blockIdx.x * blockDim.x + threadIdx.x;
    int g = C >> 2;                            // float4 groups per row
    int e = t / g, q = t - e * g;
    if (e >= E) return;
    int s = src[e], d = dst[e];
    float coef = dinv[s] * dinv[d];
    const float4 h = *((const float4*)(H + (size_t)s * C) + q);
    float* out = A + (size_t)d * C + q * 4;
    atomicAdd(out + 0, h.x * coef);
    atomicAdd(out + 1, h.y * coef);
    atomicAdd(out + 2, h.z * coef);
    atomicAdd(out + 3, h.w * coef);
}

// ---------------------------------------------------------------------------
// Attention pooling (C3 == 16)
// ---------------------------------------------------------------------------
__global__ __launch_bounds__(256)
void colsum16(const float* __restrict__ H, float* __restrict__ out, int N) {
    __shared__ float s[16];
    if (threadIdx.x < 16) s[threadIdx.x] = 0.0f;
    __syncthreads();
    float loc[16];
#pragma unroll
    for (int k = 0; k < 16; ++k) loc[k] = 0.0f;
    for (int i = blockIdx.x * blockDim.x + threadIdx.x; i < N; i += gridDim.x * blockDim.x) {
        const float* r = H + (size_t)i * 16;
#pragma unroll
        for (int k = 0; k < 16; ++k) loc[k] += r[k];
    }
#pragma unroll
    for (int k = 0; k < 16; ++k) atomicAdd(&s[k], loc[k]);
    __syncthreads();
    if (threadIdx.x < 16) atomicAdd(&out[threadIdx.x], s[threadIdx.x]);
}

__global__ void make_c(const float* __restrict__ csum, const float* __restrict__ Wa,
                       float* __restrict__ cvec, float invN) {
    int k = threadIdx.x;
    if (k < 16) {
        float acc = 0.0f;
        for (int j = 0; j < 16; ++j) acc += (csum[j] * invN) * Wa[j * 16 + k];
        cvec[k] = tanhf(acc);
    }
}

__global__ __launch_bounds__(256)
void attpool(const float* __restrict__ H, const float* __restrict__ cvec,
             float* __restrict__ e, int N) {
    __shared__ float sc[16];
    __shared__ float se[16];
    if (threadIdx.x < 16) { sc[threadIdx.x] = cvec[threadIdx.x]; se[threadIdx.x] = 0.0f; }
    __syncthreads();
    float loc[16];
#pragma unroll
    for (int k = 0; k < 16; ++k) loc[k] = 0.0f;
    for (int i = blockIdx.x * blockDim.x + threadIdx.x; i < N; i += gridDim.x * blockDim.x) {
        const float* r = H + (size_t)i * 16;
        float dot = 0.0f;
#pragma unroll
        for (int k = 0; k < 16; ++k) dot += r[k] * sc[k];
        float sg = 1.0f / (1.0f + expf(-dot));
#pragma unroll
        for (int k = 0; k < 16; ++k) loc[k] += r[k] * sg;
    }
#pragma unroll
    for (int k = 0; k < 16; ++k) atomicAdd(&se[k], loc[k]);
    __syncthreads();
    if (threadIdx.x < 16) atomicAdd(&e[threadIdx.x], se[threadIdx.x]);
}

// ---------------------------------------------------------------------------
// Final neural tensor network scoring (single block, 16 lanes of work)
// ---------------------------------------------------------------------------
__global__ void final_score(const float* __restrict__ e1, const float* __restrict__ e2,
                            const float* __restrict__ Wt, const float* __restrict__ Wb,
                            const float* __restrict__ bt, const float* __restrict__ Wfc,
                            const float* __restrict__ bfc, const float* __restrict__ Wsc,
                            const float* __restrict__ bsc, float* __restrict__ out) {
    __shared__ float scores[16];
    __shared__ float s2[16];
    int k = threadIdx.x;
    if (k < 16) {
        float bil = 0.0f;
        for (int i = 0; i < 16; ++i) {
            float a = e1[i];
            for (int j = 0; j < 16; ++j) bil += a * Wt[(i * 16 + j) * 16 + k] * e2[j];
        }
        float blk = 0.0f;
        for (int j = 0; j < 16; ++j) blk += Wb[k * 32 + j] * e1[j];
        for (int j = 0; j < 16; ++j) blk += Wb[k * 32 + 16 + j] * e2[j];
        float v = bil + blk + bt[k];
        scores[k] = v > 0.0f ? v : 0.0f;
    }
    __syncthreads();
    if (k < 16) {
        float acc = bfc[k];
        for (int j = 0; j < 16; ++j) acc += scores[j] * Wfc[j * 16 + k];
        s2[k] = tanhf(acc);
    }
    __syncthreads();
    if (k == 0) {
        float acc = bsc[0];
        for (int j = 0; j < 16; ++j) acc += s2[j] * Wsc[j];
        out[0] = 1.0f / (1.0f + expf(-acc));
    }
}

// ---------------------------------------------------------------------------
// Host orchestration
// ---------------------------------------------------------------------------
static inline int ceil_div(long long a, int b) { return (int)((a + b - 1) / b); }

static void encode_and_pool(const float* X, const int* edges, int N, int E,
                            const float* W1, const float* b1,
                            const float* W2, const float* b2,
                            const float* W3, const float* b3,
                            const float* Wa,
                            float* dinv, float* Hbuf, float* Abuf,
                            float* csum, float* cvec, float* evec,
                            hipStream_t stream) {
    const int* src = edges;
    const int* dst = edges + E;
    const int T = 256;

    // degree -> dinv
    zero_f<<<ceil_div(N, T), T, 0, stream>>>(dinv, N);
    deg_count<<<ceil_div(E, T), T, 0, stream>>>(dst, dinv, E);
    finish_dinv<<<ceil_div(N, T), T, 0, stream>>>(dinv, N);

    // layer 1: 96 -> 64, relu
    {
        int Cin = 96, Cout = 64;
        int waves = ceil_div(N, 16) * (Cout / 16);
        gemm_wmma_f32<<<ceil_div(waves, 8), 256, 0, stream>>>(X, W1, Hbuf, N, Cin, Cout);
        init_agg<<<ceil_div((long long)N * Cout, T), T, 0, stream>>>(Hbuf, dinv, b1, Abuf, N, Cout);
        edge_agg<<<ceil_div((long long)E * (Cout / 4), T), T, 0, stream>>>(src, dst, Hbuf, dinv, Abuf, E, Cout);
        relu_k<<<ceil_div((long long)N * Cout, T), T, 0, stream>>>(Abuf, N * Cout);
    }
    // layer 2: 64 -> 32, relu
    {
        int Cin = 64, Cout = 32;
        int waves = ceil_div(N, 16) * (Cout / 16);
        gemm_wmma_f32<<<ceil_div(waves, 8), 256, 0, stream>>>(Abuf, W2, Hbuf, N, Cin, Cout);
        init_agg<<<ceil_div((long long)N * Cout, T), T, 0, stream>>>(Hbuf, dinv, b2, Abuf, N, Cout);
        edge_agg<<<ceil_div((long long)E * (Cout / 4), T), T, 0, stream>>>(src, dst, Hbuf, dinv, Abuf, E, Cout);
        relu_k<<<ceil_div((long long)N * Cout, T), T, 0, stream>>>(Abuf, N * Cout);
    }
    // layer 3: 32 -> 16 (no relu)
    {
        int Cin = 32, Cout = 16;
        int waves = ceil_div(N, 16) * (Cout / 16);
        gemm_wmma_f32<<<ceil_div(waves, 8), 256, 0, stream>>>(Abuf, W3, Hbuf, N, Cin, Cout);
        init_agg<<<ceil_div((long long)N * Cout, T), T, 0, stream>>>(Hbuf, dinv, b3, Abuf, N, Cout);
        edge_agg<<<ceil_div((long long)E * (Cout / 4), T), T, 0, stream>>>(src, dst, Hbuf, dinv, Abuf, E, Cout);
    }

    // attention pooling on Abuf [N x 16]
    zero_f<<<1, 32, 0, stream>>>(csum, 16);
    zero_f<<<1, 32, 0, stream>>>(evec, 16);
    int blocks = ceil_div(N, T); if (blocks > 1024) blocks = 1024;
    colsum16<<<blocks, T, 0, stream>>>(Abuf, csum, N);
    make_c<<<1, 32, 0, stream>>>(csum, Wa, cvec, 1.0f / (float)N);
    attpool<<<blocks, T, 0, stream>>>(Abuf, cvec, evec, N);
}

extern "C" void kernel_launch(void* const* d_in, const int* in_sizes, int n_in,
                              void* d_out, int out_size, void* d_ws, size_t ws_size,
                              hipStream_t stream) {
    (void)n_in; (void)out_size; (void)ws_size;

    const float* f1  = (const float*)d_in[0];
    const float* f2  = (const float*)d_in[1];
    const int*   ed1 = (const int*)d_in[2];
    const int*   ed2 = (const int*)d_in[3];
    const float* W1  = (const float*)d_in[4];
    const float* b1  = (const float*)d_in[5];
    const float* W2  = (const float*)d_in[6];
    const float* b2  = (const float*)d_in[7];
    const float* W3  = (const float*)d_in[8];
    const float* b3  = (const float*)d_in[9];
    const float* Wa  = (const float*)d_in[10];
    const float* Wt  = (const float*)d_in[11];
    const float* Wb  = (const float*)d_in[12];
    const float* bt  = (const float*)d_in[13];
    const float* Wfc = (const float*)d_in[14];
    const float* bfc = (const float*)d_in[15];
    const float* Wsc = (const float*)d_in[16];
    const float* bsc = (const float*)d_in[17];

    int N = in_sizes[0] / 96;
    int E = in_sizes[2] / 2;

    // workspace layout (floats), 16B-aligned sections
    float* ws = (float*)d_ws;
    size_t Npad = ((size_t)N + 3) & ~(size_t)3;
    float* dinv = ws;
    float* Hbuf = dinv + Npad;
    float* Abuf = Hbuf + Npad * 64;
    float* smal = Abuf + Npad * 64;
    float* csum = smal;
    float* cvec = smal + 16;
    float* ev1  = smal + 32;
    float* ev2  = smal + 48;

    encode_and_pool(f1, ed1, N, E, W1, b1, W2, b2, W3, b3, Wa,
                    dinv, Hbuf, Abuf, csum, cvec, ev1, stream);
    encode_and_pool(f2, ed2, N, E, W1, b1, W2, b2, W3, b3, Wa,
                    dinv, Hbuf, Abuf, csum, cvec, ev2, stream);

    final_score<<<1, 32, 0, stream>>>(ev1, ev2, Wt, Wb, bt, Wfc, bfc, Wsc, bsc,
                                      (float*)d_out);
}